// TestEmbedding_68478958567962
// MI455X (gfx1250) — compile-verified
//
#include <hip/hip_runtime.h>

// Grid-embedding trilinear-per-axis interpolation for MI455X (gfx1250).
//
// Constants baked from the reference:
//   GRID_SHAPE = [76, 136, 241]  -> gs-1   = [75, 135, 240]
//   OFFSET     = [0, 76, 212]
//   CLAMP_MAX  = [75, 211, 452]
//   TABLE_ROWS = 453, EMBED_DIM = 16, out = [N, 48] f32
//
// Strategy: stage the 453x16 f32 table (29 KB) into LDS once per block using
// gfx1250 async global->LDS copies (ASYNCcnt path), then the hot loop does
// only ds_load_b128 gathers + global_store_b128 streaming writes.

namespace {

constexpr int ROWS        = 453;
constexpr int ROW_STRIDE  = 20;                 // floats: 80 B rows -> 16B aligned + bank rotation
constexpr int TAB_FLOATS  = ROWS * ROW_STRIDE;  // 9060 floats = 36,240 B of LDS
constexpr int SRC_CHUNKS  = ROWS * 4;           // 16-byte chunks in the dense source table

__launch_bounds__(256)
__global__ void grid_embed_interp(const float* __restrict__ in,
                                  const float* __restrict__ emb,
                                  float* __restrict__ out,
                                  int n)
{
    __shared__ __align__(16) float tab[TAB_FLOATS];

    // ---- Stage embedding table into LDS via CDNA5 async copy (ASYNCcnt) ----
    {
        const unsigned long long gbase = (unsigned long long)emb;
        const unsigned lbase = (unsigned)(unsigned long long)&tab[0]; // low 32 bits = LDS offset
        for (unsigned c = threadIdx.x; c < (unsigned)SRC_CHUNKS; c += blockDim.x) {
            const unsigned row  = c >> 2;
            const unsigned k    = c & 3u;
            const unsigned goff = c * 16u;                                  // dense source offset
            const unsigned loff = lbase + row * (ROW_STRIDE * 4u) + k * 16u; // padded LDS offset
            asm volatile("global_load_async_to_lds_b128 %0, %1, %2"
                         :: "v"(loff), "v"(goff), "s"(gbase)
                         : "memory");
        }
        asm volatile("s_wait_asynccnt 0" ::: "memory");
    }
    __syncthreads();

    const float gsm1[3] = {75.0f, 135.0f, 240.0f};
    const float offv[3] = {0.0f,  76.0f,  212.0f};
    const float cmax[3] = {75.0f, 211.0f, 452.0f};

    const int stride = (int)(gridDim.x * blockDim.x);
    for (int i = (int)(blockIdx.x * blockDim.x + threadIdx.x); i < n; i += stride) {
        // Prefetch next grid-stride chunk of inputs (global_prefetch_b8).
        if (i + stride < n) {
            __builtin_prefetch(in + 3 * (i + stride), 0, 0);
        }

        float p[3];
        p[0] = in[3 * i + 0];
        p[1] = in[3 * i + 1];
        p[2] = in[3 * i + 2];

        float* o = out + (long long)i * 48;

#pragma unroll
        for (int a = 0; a < 3; ++a) {
            const float c  = p[a] * gsm1[a] + offv[a];
            float f0 = floorf(c);
            float f1 = floorf(c + 1.0f);
            f0 = fminf(fmaxf(f0, offv[a]), cmax[a]);
            f1 = fminf(fmaxf(f1, offv[a]), cmax[a]);
            const float w0 = 1.0f - fabsf(f0 - c);
            const float w1 = 1.0f - fabsf(f1 - c);
            const int i0 = (int)f0;
            const int i1 = (int)f1;

            const float4* r0 = (const float4*)&tab[i0 * ROW_STRIDE]; // ds_load_b128
            const float4* r1 = (const float4*)&tab[i1 * ROW_STRIDE];

#pragma unroll
            for (int k = 0; k < 4; ++k) {
                const float4 v0 = r0[k];
                const float4 v1 = r1[k];
                float4 r;
                r.x = w0 * v0.x + w1 * v1.x;
                r.y = w0 * v0.y + w1 * v1.y;
                r.z = w0 * v0.z + w1 * v1.z;
                r.w = w0 * v0.w + w1 * v1.w;
                *(float4*)(o + a * 16 + 4 * k) = r;                  // global_store_b128
            }
        }
    }
}

} // namespace

extern "C" void kernel_launch(void* const* d_in, const int* in_sizes, int n_in,
                              void* d_out, int out_size, void* d_ws, size_t ws_size,
                              hipStream_t stream) {
    (void)n_in; (void)out_size; (void)d_ws; (void)ws_size;

    const float* inputs = (const float*)d_in[0];   // [N, 3] f32
    const float* emb    = (const float*)d_in[1];   // [453, 16] f32
    float* out          = (float*)d_out;           // [N, 48] f32

    const int n = in_sizes[0] / 3;

    int blocks = 4096;                              // grid-stride; amortize table staging
    const int max_blocks = (n + 255) / 256;
    if (blocks > max_blocks) blocks = max_blocks;
    if (blocks < 1) blocks = 1;

    grid_embed_interp<<<dim3(blocks), dim3(256), 0, stream>>>(inputs, emb, out, n);
}